// ProjectLoss_9328668967334
// MI455X (gfx1250) — compile-verified
//
#include <hip/hip_runtime.h>
#include <hip/hip_bf16.h>

#define EPS   1e-8f
#define BIG   1000000.0f

// Row length of the [H,W,64,64] grid tensor's reduction axes: 64*64 = 4096 floats = 16KB.
#define ROWLEN 4096
#define NROWS  4096
#define TPB    256

// Truncate a generic (flat) pointer to a shared variable down to its raw LDS
// byte offset: the flat LDS aperture places the offset in addr[31:0].
__device__ __forceinline__ unsigned lds_offset(const void* p) {
    return (unsigned)(unsigned long long)p;
}

// Kernel A: minD[row] = min over 4096 contiguous floats of G[row*4096 ..] + 1.0f
// One 256-thread block (8 wave32) per row. Data staged LDS-side via the CDNA5
// async global->LDS path (tracked on ASYNCcnt), then min-reduced from VGPRs.
__global__ void __launch_bounds__(TPB)
row_min_kernel(const float* __restrict__ G, float* __restrict__ minD) {
    __shared__ float tile[ROWLEN];   // 16 KB staging buffer
    __shared__ float wred[TPB / 32]; // cross-wave partials

    const int row  = blockIdx.x;
    const int t    = threadIdx.x;
    const int lane = t & 31;
    const int wid  = t >> 5;

    // Each thread owns 4 float4 chunks: byte offsets t*16 + k*4096 inside the row.
    const unsigned long long gbase =
        (unsigned long long)(G + (size_t)row * ROWLEN) + (unsigned long long)t * 16ull;
    const unsigned lbase = lds_offset(tile) + (unsigned)t * 16u;

    // Issue all four 128-bit async copies back-to-back; they ride ASYNCcnt and
    // bypass VGPRs entirely (global_load_async_to_lds_b128, GV addressing mode).
    #pragma unroll
    for (int k = 0; k < 4; ++k) {
        asm volatile("global_load_async_to_lds_b128 %0, %1, off"
                     :: "v"(lbase + (unsigned)(k * 4096)),
                        "v"(gbase + (unsigned long long)(k * 4096))
                     : "memory");
    }
    // Wait for this wave's async copies; each wave reads back only the LDS
    // bytes it loaded itself, so no workgroup barrier is required here.
    asm volatile("s_wait_asynccnt 0" ::: "memory");

    float m = 3.0e38f;
    #pragma unroll
    for (int k = 0; k < 4; ++k) {
        const float4 v = *reinterpret_cast<const float4*>(&tile[t * 4 + k * 1024]);
        m = fminf(m, fminf(fminf(v.x, v.y), fminf(v.z, v.w)));
    }

    // wave32 butterfly reduction
    m = fminf(m, __shfl_xor(m, 16, 32));
    m = fminf(m, __shfl_xor(m,  8, 32));
    m = fminf(m, __shfl_xor(m,  4, 32));
    m = fminf(m, __shfl_xor(m,  2, 32));
    m = fminf(m, __shfl_xor(m,  1, 32));

    if (lane == 0) wred[wid] = m;
    __syncthreads();

    if (t == 0) {
        float r = wred[0];
        #pragma unroll
        for (int k = 1; k < TPB / 32; ++k) r = fminf(r, wred[k]);
        minD[row] = r + 1.0f;   // dist = grid + 1.0 folded into the reduced value
    }
}

// Kernel B: elementwise epilogue over [B,H,W] (n = B*4096 elements).
//   loss         = -g*log(p+EPS) - (1-g)*log(|1-p-EPS|)
//   min_dist     = ((g + (1-g)*BIG) * minD) * p          (assoc. matches ref)
//   min_dist_inv = ((g * minD)) * (p + (1-p)*BIG)
__global__ void __launch_bounds__(TPB)
finalize_kernel(const float* __restrict__ preds,
                const float* __restrict__ gts,
                const float* __restrict__ minD,
                float* __restrict__ out, int n) {
    const int i = blockIdx.x * blockDim.x + threadIdx.x;
    if (i >= n) return;

    const int   hw = i & (NROWS - 1);
    const float p  = preds[i];
    const float g  = gts[i];
    const float d  = minD[hw];

    const float loss = -g * logf(p + EPS) - (1.0f - g) * logf(fabsf(1.0f - p - EPS));

    const float gt_th     = g + (1.0f - g) * BIG;
    const float pred_mask = p + (1.0f - p) * BIG;
    const float md  = (gt_th * d) * p;   // min(gt_th * dist * pred) by monotonicity
    const float mdi = (g * d) * pred_mask;

    out[i]          = loss;
    out[n + i]      = md;
    out[2 * n + i]  = mdi;
}

extern "C" void kernel_launch(void* const* d_in, const int* in_sizes, int n_in,
                              void* d_out, int out_size, void* d_ws, size_t ws_size,
                              hipStream_t stream) {
    const float* preds = (const float*)d_in[0];   // [B,64,64]
    const float* gts   = (const float*)d_in[1];   // [B,64,64]
    const float* grid  = (const float*)d_in[2];   // [64,64,64,64]

    float* minD = (float*)d_ws;                   // 4096 floats of scratch
    float* out  = (float*)d_out;                  // 3 * B*4096 floats

    const int n = in_sizes[0];                    // B*H*W (8192 for B=2)

    row_min_kernel<<<NROWS, TPB, 0, stream>>>(grid, minD);
    finalize_kernel<<<(n + TPB - 1) / TPB, TPB, 0, stream>>>(preds, gts, minD, out, n);
}